// SSEGCNBertClassifier_with_getAtten_25477746000192
// MI455X (gfx1250) — compile-verified
//
#include <hip/hip_runtime.h>
#include <math.h>

// Problem constants (match reference)
#define B_   32
#define L_   256
#define BD_  768
#define AD_  128
#define H_   8
#define DK_  16
#define NROW (B_ * L_)          // 8192
#define ALPHA_ 0.8f
#define NEG_BIG -1000000000.0f

typedef float v2f __attribute__((ext_vector_type(2)));
typedef float v8f __attribute__((ext_vector_type(8)));
typedef float f4  __attribute__((ext_vector_type(4)));
typedef int   i4v __attribute__((vector_size(16)));          // matches builtin param
typedef __attribute__((address_space(1))) i4v* gptr4;
typedef __attribute__((address_space(3))) i4v* lptr4;

// ---------------------------------------------------------------------------
// Async global->LDS 16-byte copy (CDNA5 GLOBAL_LOAD_ASYNC_TO_LDS_B128,
// tracked by ASYNCcnt).  Flat LDS pointer low 32 bits == LDS byte offset
// (aperture rule), so we build the addrspace(3) pointer by truncation.
// ---------------------------------------------------------------------------
static __device__ __forceinline__ void async_cp16(const float* g, float* l) {
#if __has_builtin(__builtin_amdgcn_global_load_async_to_lds_b128)
  __builtin_amdgcn_global_load_async_to_lds_b128(
      (gptr4)(unsigned long long)(size_t)g,
      (lptr4)(unsigned)(size_t)l, 0, 0);
#else
  *(f4*)l = *(const f4*)g;      // synchronous vectorized fallback
#endif
}

static __device__ __forceinline__ void wait_async0() {
#if __has_builtin(__builtin_amdgcn_s_wait_asynccnt)
  __builtin_amdgcn_s_wait_asynccnt(0);
#else
  asm volatile("s_wait_asynccnt 0x0" ::: "memory");
#endif
}

// ---------------------------------------------------------------------------
// 1) LayerNorm over BD=768 (one block per row, 256 threads)
// ---------------------------------------------------------------------------
__global__ __launch_bounds__(256) void k_ln_bd(const float* __restrict__ xin,
                                               const float* __restrict__ w,
                                               const float* __restrict__ b,
                                               float* __restrict__ xout) {
  int row = blockIdx.x;
  const float* xr = xin + (size_t)row * BD_;
  float* yr = xout + (size_t)row * BD_;
  __shared__ float red[256];
  int t = threadIdx.x;

  float s = 0.f;
  for (int c = t; c < BD_; c += 256) s += xr[c];
  red[t] = s; __syncthreads();
  for (int off = 128; off > 0; off >>= 1) { if (t < off) red[t] += red[t + off]; __syncthreads(); }
  float mean = red[0] / (float)BD_; __syncthreads();

  float v = 0.f;
  for (int c = t; c < BD_; c += 256) { float d = xr[c] - mean; v += d * d; }
  red[t] = v; __syncthreads();
  for (int off = 128; off > 0; off >>= 1) { if (t < off) red[t] += red[t + off]; __syncthreads(); }
  float inv = rsqrtf(red[0] / (float)BD_ + 1e-5f);

  for (int c = t; c < BD_; c += 256) yr[c] = (xr[c] - mean) * inv * w[c] + b[c];
}

// ---------------------------------------------------------------------------
// 2) WMMA GEMM with async double-buffered LDS staging:
//    C[M][128] = A[M][K] @ W[128][K]^T + bias[128]
//    Block = 16 rows x 128 cols, 8 waves each own a 16x16 WMMA tile.
//    While chunk k is consumed by v_wmma_f32_16x16x4_f32, chunk k+1 streams
//    into the other LDS buffer via GLOBAL_LOAD_ASYNC_TO_LDS_B128.
//    Row stride 36 floats: 16B-aligned b128 stores + conflict-free frag reads.
// ---------------------------------------------------------------------------
__global__ __launch_bounds__(256) void k_gemm_wmma(const float* __restrict__ A,
                                                   const float* __restrict__ W,
                                                   const float* __restrict__ bias,
                                                   float* __restrict__ C,
                                                   int M, int K) {
  const int KC = 32;
  __shared__ __align__(16) float As[2][16][36];
  __shared__ __align__(16) float Ws[2][128][36];

  int t = threadIdx.x;
  int wave = t >> 5, lane = t & 31;
  int hh = lane >> 4, lr = lane & 15;
  int m0 = blockIdx.x * 16;
  int nChunks = K / KC;

#define GEMM_STAGE(bufi, kbase)                                              \
  do {                                                                       \
    if (t < 128) {                                                           \
      int r = t >> 3, c4 = (t & 7) * 4;                                      \
      async_cp16(&A[(size_t)(m0 + r) * K + (kbase) + c4], &As[bufi][r][c4]); \
    }                                                                        \
    for (int idx = t; idx < 1024; idx += 256) {                              \
      int r = idx >> 3, c4 = (idx & 7) * 4;                                  \
      async_cp16(&W[(size_t)r * K + (kbase) + c4], &Ws[bufi][r][c4]);        \
    }                                                                        \
  } while (0)

  GEMM_STAGE(0, 0);
  wait_async0();
  __syncthreads();

  v8f acc = {};
  for (int ch = 0; ch < nChunks; ++ch) {
    int cur = ch & 1;
    if (ch + 1 < nChunks) GEMM_STAGE(cur ^ 1, (ch + 1) * KC);
#pragma unroll
    for (int kk = 0; kk < KC; kk += 4) {
      v2f a, bf;
      a.x  = As[cur][lr][kk + 2 * hh + 0];
      a.y  = As[cur][lr][kk + 2 * hh + 1];
      bf.x = Ws[cur][wave * 16 + lr][kk + 2 * hh + 0];   // B[k][n] = W[n][k]
      bf.y = Ws[cur][wave * 16 + lr][kk + 2 * hh + 1];
      acc = __builtin_amdgcn_wmma_f32_16x16x4_f32(false, a, false, bf,
                                                  (short)0, acc, false, false);
    }
    wait_async0();
    __syncthreads();
  }
#undef GEMM_STAGE

  int col = wave * 16 + lr;
  float bb = bias[col];
#pragma unroll
  for (int r = 0; r < 8; ++r) {
    int row = m0 + r + 8 * hh;
    C[(size_t)row * AD_ + col] = acc[r] + bb;
  }
}

// ---------------------------------------------------------------------------
// 3) aspect[b][d] = sum_l x[b,l,d]*amask[b,l] / sum_l amask[b,l]
// ---------------------------------------------------------------------------
__global__ __launch_bounds__(128) void k_aspect(const float* __restrict__ x,
                                                const float* __restrict__ amask,
                                                float* __restrict__ aspect) {
  int b = blockIdx.x, d = threadIdx.x;
  float s = 0.f, wn = 0.f;
  for (int l = 0; l < L_; ++l) {
    float m = amask[b * L_ + l];
    s  += x[((size_t)b * L_ + l) * AD_ + d] * m;
    wn += m;
  }
  aspect[b * AD_ + d] = s / wn;
}

// ---------------------------------------------------------------------------
// 4) Sinkhorn pi per (b,h).  nu = softmax over a singleton = 1, so
//    v = 1/(sum_i K_i u_i + eps).  50 fixed iterations.
// ---------------------------------------------------------------------------
__global__ __launch_bounds__(256) void k_pi(const float* __restrict__ x,
                                            const float* __restrict__ aspect,
                                            const float* __restrict__ smask,
                                            const float* __restrict__ mu_w,
                                            const float* __restrict__ mu_b,
                                            float* __restrict__ pi) {
  int bh = blockIdx.x;               // b*H + h
  int b = bh >> 3, h = bh & 7;
  int i = threadIdx.x;               // 0..255 == L
  __shared__ float red[256];

  float s[DK_], an[DK_];
#pragma unroll
  for (int d = 0; d < DK_; ++d) {
    s[d]  = x[((size_t)b * L_ + i) * AD_ + h * DK_ + d];
    an[d] = aspect[b * AD_ + h * DK_ + d];
  }
  float sn2 = 0.f, an2 = 0.f, dots = 0.f, ml = 0.f;
#pragma unroll
  for (int d = 0; d < DK_; ++d) {
    sn2 += s[d] * s[d];
    an2 += an[d] * an[d];
    dots += s[d] * an[d];
    ml  += s[d] * mu_w[d];
  }
  float cost = 1.f - dots / (fmaxf(sqrtf(sn2), 1e-12f) * fmaxf(sqrtf(an2), 1e-12f));
  float Kv = expf(-cost);            // OT_EPS = 1
  float mu_logit = ml + mu_b[0];
  if (smask[b * L_ + i] == 0.f) mu_logit = NEG_BIG;

  // masked softmax over i
  red[i] = mu_logit; __syncthreads();
  for (int off = 128; off > 0; off >>= 1) { if (i < off) red[i] = fmaxf(red[i], red[i + off]); __syncthreads(); }
  float mx = red[0]; __syncthreads();
  float e = expf(mu_logit - mx);
  red[i] = e; __syncthreads();
  for (int off = 128; off > 0; off >>= 1) { if (i < off) red[i] += red[i + off]; __syncthreads(); }
  float mu = e / red[0]; __syncthreads();

  float u = 1.f, v = 1.f;
  for (int it = 0; it < 50; ++it) {
    u = mu / (Kv * v + 1e-8f);
    red[i] = Kv * u; __syncthreads();
    for (int off = 128; off > 0; off >>= 1) { if (i < off) red[i] += red[i + off]; __syncthreads(); }
    v = 1.f / (red[0] + 1e-8f);
    __syncthreads();
  }
  pi[(size_t)bh * L_ + i] = u * Kv * v;
}

// ---------------------------------------------------------------------------
// 5) Fused attention per (b, h, 16-query tile):
//    async-stage Q/K/V -> WMMA QK^T -> mask/+short_mask -> softmax (LDS)
//    -> WMMA attn@V (K split over 8 waves, LDS reduce)
//    -> alpha*sent + (1-alpha)*pi*seq_h + bias_m
// ---------------------------------------------------------------------------
__global__ __launch_bounds__(256) void k_attn(const float* __restrict__ q,
                                              const float* __restrict__ kbuf,
                                              const float* __restrict__ x,
                                              const float* __restrict__ pi,
                                              const float* __restrict__ smask,
                                              const float* __restrict__ short_mask,
                                              const float* __restrict__ bias_m,
                                              float* __restrict__ out) {
  int blk = blockIdx.x;
  int it = blk & 15;
  int bh = blk >> 4;
  int b = bh >> 3, h = bh & 7;
  int i0 = it * 16;
  int t = threadIdx.x;
  int wave = t >> 5, lane = t & 31, hh = lane >> 4, lr = lane & 15;

  __shared__ __align__(16) float Qs[16][20];
  __shared__ __align__(16) float Kh[256][20];
  __shared__ __align__(16) float Vh[256][16];
  __shared__ float S[16][257];
  __shared__ float Sp[8][16][16];
  __shared__ float red[16][17];

  // async stage Q tile (64 x 16B), K head (1024 x 16B), V head (1024 x 16B)
  if (t < 64) {
    int i = t >> 2, c4 = (t & 3) * 4;
    async_cp16(&q[((size_t)b * L_ + i0 + i) * AD_ + h * DK_ + c4], &Qs[i][c4]);
  }
  for (int idx = t; idx < 1024; idx += 256) {
    int j = idx >> 2, c4 = (idx & 3) * 4;
    async_cp16(&kbuf[((size_t)b * L_ + j) * AD_ + h * DK_ + c4], &Kh[j][c4]);
    async_cp16(&x[((size_t)b * L_ + j) * AD_ + h * DK_ + c4], &Vh[j][c4]);
  }
  // prefetch this block's 16KB short_mask tile (one 64B line per thread)
  __builtin_prefetch(short_mask + (((size_t)bh) * L_ + i0) * L_ + t * 16, 0, 0);
  wait_async0();
  __syncthreads();

  const float scale = 0.25f;          // 1/sqrt(DK=16)

  // ---- scores: each wave does key tiles jt = wave, wave+8 ----
  for (int rep = 0; rep < 2; ++rep) {
    int j0 = (wave + rep * 8) * 16;
    v8f acc = {};
#pragma unroll
    for (int kk = 0; kk < DK_; kk += 4) {
      v2f a, bf;
      a.x  = Qs[lr][kk + 2 * hh];      a.y  = Qs[lr][kk + 2 * hh + 1];
      bf.x = Kh[j0 + lr][kk + 2 * hh]; bf.y = Kh[j0 + lr][kk + 2 * hh + 1];
      acc = __builtin_amdgcn_wmma_f32_16x16x4_f32(false, a, false, bf,
                                                  (short)0, acc, false, false);
    }
#pragma unroll
    for (int r = 0; r < 8; ++r) {
      int il = r + 8 * hh;
      int jg = j0 + lr;
      float val = acc[r] * scale;
      if (smask[b * L_ + jg] == 0.f) val = NEG_BIG;
      val += short_mask[(((size_t)bh) * L_ + (i0 + il)) * L_ + jg];
      S[il][jg] = val;
    }
  }
  __syncthreads();

  // ---- row softmax over 256 keys: 16 threads per row ----
  {
    int row = t >> 4, c0 = t & 15;
    float mx = -INFINITY;
    for (int c = c0; c < 256; c += 16) mx = fmaxf(mx, S[row][c]);
    red[row][c0] = mx; __syncthreads();
    for (int off = 8; off > 0; off >>= 1) { if (c0 < off) red[row][c0] = fmaxf(red[row][c0], red[row][c0 + off]); __syncthreads(); }
    float rmax = red[row][0]; __syncthreads();
    float ssum = 0.f;
    for (int c = c0; c < 256; c += 16) { float e = expf(S[row][c] - rmax); S[row][c] = e; ssum += e; }
    red[row][c0] = ssum; __syncthreads();
    for (int off = 8; off > 0; off >>= 1) { if (c0 < off) red[row][c0] += red[row][c0 + off]; __syncthreads(); }
    float inv = 1.f / red[row][0]; __syncthreads();
    for (int c = c0; c < 256; c += 16) S[row][c] *= inv;
  }
  __syncthreads();

  // ---- sent = attn(16x256) @ V(256x16); K split across waves ----
  {
    v8f acc = {};
    int kbase = wave * 32;
#pragma unroll
    for (int kk = 0; kk < 32; kk += 4) {
      int k = kbase + kk + 2 * hh;
      v2f a, bf;
      a.x  = S[lr][k];     a.y  = S[lr][k + 1];
      bf.x = Vh[k][lr];    bf.y = Vh[k + 1][lr];   // B[k][n] = Vh[k][n]
      acc = __builtin_amdgcn_wmma_f32_16x16x4_f32(false, a, false, bf,
                                                  (short)0, acc, false, false);
    }
#pragma unroll
    for (int r = 0; r < 8; ++r) Sp[wave][r + 8 * hh][lr] = acc[r];
  }
  __syncthreads();

  // ---- reduce partials, combine with pi, write (B,L,AD) slice ----
  {
    int il = t >> 4, d = t & 15;
    float s = 0.f;
#pragma unroll
    for (int w2 = 0; w2 < 8; ++w2) s += Sp[w2][il][d];
    int ig = i0 + il;
    float p = pi[(size_t)bh * L_ + ig];
    float o = ALPHA_ * s + (1.f - ALPHA_) * p * Vh[ig][d] + bias_m[0];
    out[((size_t)b * L_ + ig) * AD_ + h * DK_ + d] = o;
  }
}

// ---------------------------------------------------------------------------
// 6) x = LN(x + out) over AD=128, in place (one block per row)
// ---------------------------------------------------------------------------
__global__ __launch_bounds__(128) void k_ln_res(float* __restrict__ x,
                                                const float* __restrict__ out,
                                                const float* __restrict__ w,
                                                const float* __restrict__ b2) {
  int row = blockIdx.x;
  int t = threadIdx.x;
  __shared__ float red[128];
  size_t base = (size_t)row * AD_;
  float y = x[base + t] + out[base + t];
  red[t] = y; __syncthreads();
  for (int off = 64; off > 0; off >>= 1) { if (t < off) red[t] += red[t + off]; __syncthreads(); }
  float mean = red[0] / (float)AD_; __syncthreads();
  float d = y - mean;
  red[t] = d * d; __syncthreads();
  for (int off = 64; off > 0; off >>= 1) { if (t < off) red[t] += red[t + off]; __syncthreads(); }
  float inv = rsqrtf(red[0] / (float)AD_ + 1e-5f);
  x[base + t] = d * inv * w[t] + b2[t];
}

// ---------------------------------------------------------------------------
// 7) relu -> masked pool -> classifier (B,3)
// ---------------------------------------------------------------------------
__global__ __launch_bounds__(128) void k_final(const float* __restrict__ x,
                                               const float* __restrict__ amask,
                                               const float* __restrict__ cls_w,
                                               const float* __restrict__ cls_b,
                                               float* __restrict__ outp) {
  int b = blockIdx.x, d = threadIdx.x;
  __shared__ float pooled[128];
  float s = 0.f, wn = 0.f;
  for (int l = 0; l < L_; ++l) {
    float m = amask[b * L_ + l];
    wn += m;
    s  += fmaxf(x[((size_t)b * L_ + l) * AD_ + d], 0.f) * m;
  }
  pooled[d] = s / wn; __syncthreads();
  if (d < 3) {
    float acc = cls_b[d];
    for (int c = 0; c < AD_; ++c) acc += pooled[c] * cls_w[d * AD_ + c];
    outp[b * 3 + d] = acc;
  }
}

// ---------------------------------------------------------------------------
extern "C" void kernel_launch(void* const* d_in, const int* in_sizes, int n_in,
                              void* d_out, int out_size, void* d_ws, size_t ws_size,
                              hipStream_t stream) {
  const float* seq        = (const float*)d_in[0];
  const float* src_mask   = (const float*)d_in[1];
  const float* amask      = (const float*)d_in[2];
  const float* short_mask = (const float*)d_in[3];
  const float* bln_w      = (const float*)d_in[4];
  const float* bln_b      = (const float*)d_in[5];
  const float* Wxx_w      = (const float*)d_in[6];
  const float* Wxx_b      = (const float*)d_in[7];
  const float* Wq_w       = (const float*)d_in[8];
  const float* Wq_b       = (const float*)d_in[9];
  const float* Wk_w       = (const float*)d_in[10];
  const float* Wk_b       = (const float*)d_in[11];
  const float* bias_m     = (const float*)d_in[12];
  const float* mu_w       = (const float*)d_in[13];
  const float* mu_b       = (const float*)d_in[14];
  const float* aln_w      = (const float*)d_in[15];
  const float* aln_b      = (const float*)d_in[16];
  const float* cls_w      = (const float*)d_in[17];
  const float* cls_b      = (const float*)d_in[18];

  float* ws = (float*)d_ws;
  size_t off = 0;
  float* xln    = ws + off; off += (size_t)NROW * BD_;
  float* x      = ws + off; off += (size_t)NROW * AD_;
  float* qb     = ws + off; off += (size_t)NROW * AD_;
  float* kb     = ws + off; off += (size_t)NROW * AD_;
  float* outb   = ws + off; off += (size_t)NROW * AD_;
  float* aspect = ws + off; off += (size_t)B_ * AD_;
  float* pib    = ws + off; off += (size_t)B_ * H_ * L_;
  (void)in_sizes; (void)n_in; (void)out_size; (void)ws_size;

  k_ln_bd<<<NROW, 256, 0, stream>>>(seq, bln_w, bln_b, xln);
  k_gemm_wmma<<<NROW / 16, 256, 0, stream>>>(xln, Wxx_w, Wxx_b, x, NROW, BD_);
  k_aspect<<<B_, AD_, 0, stream>>>(x, amask, aspect);

  for (int layer = 0; layer < 2; ++layer) {
    k_gemm_wmma<<<NROW / 16, 256, 0, stream>>>(x, Wq_w, Wq_b, qb, NROW, AD_);
    k_gemm_wmma<<<NROW / 16, 256, 0, stream>>>(x, Wk_w, Wk_b, kb, NROW, AD_);
    k_pi<<<B_ * H_, L_, 0, stream>>>(x, aspect, src_mask, mu_w, mu_b, pib);
    k_attn<<<B_ * H_ * (L_ / 16), 256, 0, stream>>>(qb, kb, x, pib, src_mask,
                                                    short_mask, bias_m, outb);
    k_ln_res<<<NROW, AD_, 0, stream>>>(x, outb, aln_w, aln_b);
  }
  k_final<<<B_, AD_, 0, stream>>>(x, amask, cls_w, cls_b, (float*)d_out);
}